// GIN_420906795778
// MI455X (gfx1250) — compile-verified
//
#include <hip/hip_runtime.h>

#define HID   64
#define NNODE 50000
#define NEDGE 1600000
#define NTILE 3125           // 50000 / 16

#define FXSCALE     1048576.0f            // 2^20 fixed-point scale
#define FXINVSCALE  (1.0f / 1048576.0f)

typedef float v2f __attribute__((ext_vector_type(2)));
typedef float v4f __attribute__((ext_vector_type(4)));
typedef float v8f __attribute__((ext_vector_type(8)));
typedef int   v2i __attribute__((ext_vector_type(2)));
typedef int   v4i __attribute__((ext_vector_type(4)));

// ---------------------------------------------------------------- zero agg (int32 accumulator)
__global__ __launch_bounds__(256) void gin_zero(v4i* __restrict__ p, int n4) {
    int i = blockIdx.x * 256 + threadIdx.x;
    if (i < n4) p[i] = (v4i){0, 0, 0, 0};
}

// ------------------------------------------------- edge scatter: agg[dst] += fx(h[src] + e)
// 16 threads per edge, each thread handles a float4 chunk (coalesced 512B/wave).
// edge_feat is a one-shot 410MB stream -> non-temporal so node_feat/agg stay L2-resident.
// Fixed-point int32 atomics => bitwise-deterministic accumulation.
__global__ __launch_bounds__(256) void gin_edge(const float* __restrict__ node_feat,
                                                const float* __restrict__ edge_feat,
                                                const int*   __restrict__ src,
                                                const int*   __restrict__ dst,
                                                int*         __restrict__ agg) {
    unsigned tid = blockIdx.x * 256u + threadIdx.x;
    unsigned e   = tid >> 4;
    if (e >= NEDGE) return;
    int c = (int)(tid & 15u) * 4;

    int s = src[e];
    int d = dst[e];

    const v4f ef = __builtin_nontemporal_load(
                       (const v4f*)(edge_feat + (size_t)e * HID + c));
    const v4f nf = *(const v4f*)(node_feat + (size_t)s * HID + c);

    int* ap = agg + (size_t)d * HID + c;
    atomicAdd(ap + 0, __float2int_rn((ef.x + nf.x) * FXSCALE));
    atomicAdd(ap + 1, __float2int_rn((ef.y + nf.y) * FXSCALE));
    atomicAdd(ap + 2, __float2int_rn((ef.z + nf.z) * FXSCALE));
    atomicAdd(ap + 3, __float2int_rn((ef.w + nf.w) * FXSCALE));
}

// ------------------------------------------------- MLP: out = relu(agg@W1+b1)@W2+b2
// 8 waves / block, 1 wave handles one 16-row node tile using fp32 WMMA 16x16x4.
__global__ __launch_bounds__(256) void gin_mlp(const int*   __restrict__ agg,  // fx-int32 [50000][64]
                                               const float* __restrict__ W1,   // [64][128]
                                               const float* __restrict__ b1,   // [128]
                                               const float* __restrict__ W2,   // [128][64]
                                               const float* __restrict__ b2,   // [64]
                                               float*       __restrict__ out) {
    __shared__ float lds_h[8 * 16 * 128];   // per-wave 16x128 hidden tile (64 KB)

    const int wave = threadIdx.x >> 5;
    const int lane = threadIdx.x & 31;
    const int half = lane >> 4;       // 0: lanes 0-15, 1: lanes 16-31
    const int l16  = lane & 15;

    int tile = blockIdx.x * 8 + wave;
    if (tile >= NTILE) tile = NTILE - 1;        // clamped waves redo last tile (same data, benign)
    const int row_base = tile * 16;

    float* hws = lds_h + wave * (16 * 128);

    // ---------------- layer 1: h = relu(A[16x64] @ W1[64x128] + b1) ----------------
    // A fragment layout (f32 16x4): lane<16 -> M=lane, K={k0,k0+1}; lane>=16 -> K={k0+2,k0+3}
    v2f afrag[16];
#pragma unroll
    for (int kc = 0; kc < 16; ++kc) {
        int k0 = kc * 4 + half * 2;
        v2i ai = *(const v2i*)(agg + (size_t)(row_base + l16) * HID + k0);
        v2f a;
        a.x = (float)ai.x * FXINVSCALE;
        a.y = (float)ai.y * FXINVSCALE;
        afrag[kc] = a;
    }

#pragma unroll
    for (int nt = 0; nt < 8; ++nt) {
        const int n = nt * 16 + l16;
        v8f acc = {};
#pragma unroll
        for (int kc = 0; kc < 16; ++kc) {
            int k0 = kc * 4 + half * 2;
            v2f b;
            b.x = W1[(k0 + 0) * 128 + n];
            b.y = W1[(k0 + 1) * 128 + n];
            acc = __builtin_amdgcn_wmma_f32_16x16x4_f32(
                    false, afrag[kc], false, b, (short)0, acc, false, false);
        }
        const float bias = b1[n];
#pragma unroll
        for (int r = 0; r < 8; ++r) {
            float h = acc[r] + bias;
            h = h > 0.f ? h : 0.f;                    // ReLU
            hws[(r + half * 8) * 128 + (nt * 16 + l16)] = h;
        }
    }

    __syncthreads();

    // ---------------- layer 2: out = h[16x128] @ W2[128x64] + b2 ----------------
#pragma unroll
    for (int nt = 0; nt < 4; ++nt) {
        const int n = nt * 16 + l16;
        v8f acc = {};
#pragma unroll
        for (int kc = 0; kc < 32; ++kc) {
            int k0 = kc * 4 + half * 2;
            v2f a = *(const v2f*)(hws + l16 * 128 + k0);
            v2f b;
            b.x = W2[(k0 + 0) * 64 + n];
            b.y = W2[(k0 + 1) * 64 + n];
            acc = __builtin_amdgcn_wmma_f32_16x16x4_f32(
                    false, a, false, b, (short)0, acc, false, false);
        }
        const float bias = b2[n];
#pragma unroll
        for (int r = 0; r < 8; ++r) {
            out[(size_t)(row_base + r + half * 8) * HID + (nt * 16 + l16)] = acc[r] + bias;
        }
    }
}

// ---------------------------------------------------------------- launcher
extern "C" void kernel_launch(void* const* d_in, const int* in_sizes, int n_in,
                              void* d_out, int out_size, void* d_ws, size_t ws_size,
                              hipStream_t stream) {
    const float* node_feat = (const float*)d_in[0];
    const float* edge_feat = (const float*)d_in[1];
    const int*   src       = (const int*)  d_in[2];
    const int*   dst       = (const int*)  d_in[3];
    const float* W1        = (const float*)d_in[4];
    const float* b1        = (const float*)d_in[5];
    const float* W2        = (const float*)d_in[6];
    const float* b2        = (const float*)d_in[7];
    float*       out       = (float*)d_out;
    int*         agg       = (int*)d_ws;          // 50000*64 int32 = 12.8 MB scratch

    // 1) zero the fixed-point accumulator
    {
        int n4 = NNODE * HID / 4;
        gin_zero<<<(n4 + 255) / 256, 256, 0, stream>>>((v4i*)agg, n4);
    }
    // 2) edge scatter with deterministic int32 atomics (L2-resident accumulator)
    {
        long total = (long)NEDGE * 16;
        int blocks = (int)((total + 255) / 256);
        gin_edge<<<blocks, 256, 0, stream>>>(node_feat, edge_feat, src, dst, agg);
    }
    // 3) WMMA fp32 MLP
    {
        int blocks = (NTILE + 7) / 8;
        gin_mlp<<<blocks, 256, 0, stream>>>(agg, W1, b1, W2, b2, out);
    }
}